// StructLoss_41558103556725
// MI455X (gfx1250) — compile-verified
//
#include <hip/hip_runtime.h>
#include <math.h>

typedef __attribute__((ext_vector_type(2))) float v2f;
typedef __attribute__((ext_vector_type(8))) float v8f;

#define HH 512
#define WW 512
#define RPB 32          // rows per block
#define TPB 128         // 4 wave32; each thread owns 4 columns (float4)
#define STRIDE 520      // floats per LDS row buffer (keeps [4t..4t+5] 16B-aligned)

// Sum p over all 32 lanes of a wave using one V_WMMA_F32_16X16X4_F32.
// A (16x4 f32): vgpr0 lanes0-15 -> A[m][0], lanes16-31 -> A[m][2]; vgpr1 -> K=1,3.
// a0=p, a1=0  =>  A[m][0]=p[m], A[m][2]=p[m+16].  B = all ones (layout-free).
// D[m][n] = p[m]+p[m+16];  D vgpr v: lanes0-15 = q[v], lanes16-31 = q[v+8].
// Sum the 8 D vgprs, then one SWAPX16 ds_swizzle finishes the 32-lane sum.
__device__ __forceinline__ float wave_sum32_wmma(float p) {
  v2f a; a[0] = p;    a[1] = 0.0f;
  v2f b; b[0] = 1.0f; b[1] = 1.0f;
  v8f c = {};
  c = __builtin_amdgcn_wmma_f32_16x16x4_f32(false, a, false, b, (short)0, c,
                                            false, false);
  float s = ((c[0] + c[1]) + (c[2] + c[3])) + ((c[4] + c[5]) + (c[6] + c[7]));
  s += __int_as_float(__builtin_amdgcn_ds_swizzle(__float_as_int(s), 0x401f));
  return s;
}

__global__ __launch_bounds__(TPB)
void grad_l1_partial(const float* __restrict__ outp,
                     const float* __restrict__ labp,
                     float* __restrict__ partial) {
  // 4 rotating d-row buffers; halo col -1 at [0], col WW at [WW+1]
  __shared__ __align__(16) float sh[4 * STRIDE];
  __shared__ float wsum[TPB / 32];

  const int t     = threadIdx.x;
  const int plane = blockIdx.x >> 4;            // 16 chunks per plane
  const int r0    = (blockIdx.x & 15) * RPB;
  const int c4    = 4 * t;                      // first owned column

  const float* o = outp + (size_t)plane * (HH * WW);
  const float* l = labp + (size_t)plane * (HH * WW);

  auto gload = [&](const float* base, int r) -> float4 {
    return *reinterpret_cast<const float4*>(base + (size_t)r * WW + c4);
  };
  auto store_d = [&](int bi, float4 d) {
    float* buf = sh + bi * STRIDE + 1 + c4;
    buf[0] = d.x; buf[1] = d.y; buf[2] = d.z; buf[3] = d.w;
  };
  const float4 fz = make_float4(0.f, 0.f, 0.f, 0.f);

  // Zero halos once for all 4 buffers.
  if (t < 4)       sh[t * STRIDE] = 0.0f;
  else if (t < 8)  sh[(t - 4) * STRIDE + (WW + 1)] = 0.0f;

  // Preload LDS rows r0-1 and r0  (row r -> buffer r&3; row -1 -> 3 = (r0+3)&3).
  if (r0 > 0) {
    const float4 a = gload(o, r0 - 1), b = gload(l, r0 - 1);
    store_d((r0 + 3) & 3, make_float4(a.x - b.x, a.y - b.y, a.z - b.z, a.w - b.w));
  } else {
    store_d(3, fz);
  }
  {
    const float4 a = gload(o, r0), b = gload(l, r0);
    store_d(r0 & 3, make_float4(a.x - b.x, a.y - b.y, a.z - b.z, a.w - b.w));
  }

  // Register prefetch of row r0+1 (always valid: r0+1 <= 481).
  float4 no = gload(o, r0 + 1);
  float4 nl = gload(l, r0 + 1);

  float acc = 0.0f;

#pragma unroll 4
  for (int i = r0; i < r0 + RPB; ++i) {
    // Stage row i+1 (from last iteration's register prefetch) into LDS.
    store_d((i + 1) & 3, make_float4(no.x - nl.x, no.y - nl.y,
                                     no.z - nl.z, no.w - nl.w));
    // Issue prefetch of row i+2; it has a full iteration to land.
    const int rn = i + 2;
    if (rn < HH) { no = gload(o, rn); nl = gload(l, rn); }
    else         { no = fz;           nl = fz;           }

    __syncthreads();   // single barrier/iter: 4-buffer rotation makes WAR safe

    const float* A  = sh + ((i + 3) & 3) * STRIDE + c4;  // row i-1, window cols c4-1..c4+4
    const float* Bm = sh + ( i      & 3) * STRIDE + c4;  // row i
    const float* Cr = sh + ((i + 1) & 3) * STRIDE + c4;  // row i+1

    const float4 a0 = *reinterpret_cast<const float4*>(A);
    const float2 a1 = *reinterpret_cast<const float2*>(A + 4);
    const float4 b0 = *reinterpret_cast<const float4*>(Bm);
    const float2 b1 = *reinterpret_cast<const float2*>(Bm + 4);
    const float4 c0 = *reinterpret_cast<const float4*>(Cr);
    const float2 c1 = *reinterpret_cast<const float2*>(Cr + 4);

    const float av[6] = {a0.x, a0.y, a0.z, a0.w, a1.x, a1.y};
    const float bv[6] = {b0.x, b0.y, b0.z, b0.w, b1.x, b1.y};
    const float cv[6] = {c0.x, c0.y, c0.z, c0.w, c1.x, c1.y};

#pragma unroll
    for (int k = 0; k < 4; ++k) {
      const float gy = av[k + 1] - cv[k + 1];   // d[i-1,j]   - d[i+1,j]
      const float gx = bv[k]     - bv[k + 2];   // d[i,j-1]   - d[i,j+1]
      const float gD = av[k]     - cv[k + 2];   // d[i-1,j-1] - d[i+1,j+1]
      const float gd = av[k + 2] - cv[k];       // d[i-1,j+1] - d[i+1,j-1]
      acc += (fabsf(gy) + fabsf(gx)) + (fabsf(gD) + fabsf(gd));
    }
  }

  // Wave reduction via WMMA (no divergence here -> EXEC all ones).
  const float wtot = wave_sum32_wmma(acc);
  const int wid  = t >> 5;
  const int lane = t & 31;
  if (lane == 0) wsum[wid] = wtot;
  __syncthreads();
  if (t == 0)
    partial[blockIdx.x] = (wsum[0] + wsum[1]) + (wsum[2] + wsum[3]);
}

__global__ __launch_bounds__(256)
void finalize_kernel(const float* __restrict__ partial, int n,
                     float* __restrict__ outv, double inv_count) {
  __shared__ double red[256];
  double s = 0.0;
  for (int i = threadIdx.x; i < n; i += 256) s += (double)partial[i];
  red[threadIdx.x] = s;
  __syncthreads();
  for (int off = 128; off > 0; off >>= 1) {
    if ((int)threadIdx.x < off) red[threadIdx.x] += red[threadIdx.x + off];
    __syncthreads();
  }
  if (threadIdx.x == 0) {
    const double mean = red[0] * inv_count;
    outv[0] = (float)sqrt(mean + 1e-16);
  }
}

extern "C" void kernel_launch(void* const* d_in, const int* in_sizes, int n_in,
                              void* d_out, int out_size, void* d_ws, size_t ws_size,
                              hipStream_t stream) {
  const float* outp = (const float*)d_in[0];
  const float* labp = (const float*)d_in[1];
  float* ws  = (float*)d_ws;
  float* res = (float*)d_out;

  const int total  = in_sizes[0];                  // B*C*H*W = 25,165,824
  const int planes = total / (HH * WW);            // 96
  const int nBlocks = planes * (HH / RPB);         // 96 * 16 = 1536

  grad_l1_partial<<<nBlocks, TPB, 0, stream>>>(outp, labp, ws);

  const double inv_count = 1.0 / (4.0 * (double)total);  // mean over [B,4C,H,W]
  finalize_kernel<<<1, 256, 0, stream>>>(ws, nBlocks, res, inv_count);
}